// Patch_Encoding_76673756168421
// MI455X (gfx1250) — compile-verified
//
#include <hip/hip_runtime.h>
#include <math.h>

// ---------------------------------------------------------------------------
// Patch-Encoding + 2-layer Mamba pipeline for MI455X (gfx1250, wave32, WMMA)
// fp32 path throughout using V_WMMA_F32_16X16X4_F32 (memory-bound GEMM shapes,
// so fp32 matrix math is essentially free vs bf16 and keeps full accuracy).
// GEMM A-tiles are staged with GLOBAL_LOAD_ASYNC_TO_LDS_B128 (ASYNCcnt) and
// double-buffered in LDS to overlap HBM/L2 latency with the WMMA chain.
// ---------------------------------------------------------------------------

typedef float v2f __attribute__((ext_vector_type(2)));
typedef float v8f __attribute__((ext_vector_type(8)));
typedef __attribute__((address_space(3))) float as3f;

#define EPS_RMS 1.1920928955078125e-07f

__device__ __forceinline__ v8f wmma4(v2f a, v2f b, v8f c) {
  // D(16x16,f32) = A(16x4,f32) x B(4x16,f32) + C
  return __builtin_amdgcn_wmma_f32_16x16x4_f32(false, a, false, b, (short)0, c,
                                               false, false);
}

// Async 16B global -> LDS copy, per-lane, tracked with ASYNCcnt.
__device__ __forceinline__ void async_ld_b128(unsigned lds_off, const float* g) {
  asm volatile("global_load_async_to_lds_b128 %0, %1, off"
               :: "v"(lds_off), "v"(g)
               : "memory");
}

// ---------------------------------------------------------------------------
// K0: zero the scatter buffer (float4 grid-stride)
// ---------------------------------------------------------------------------
__global__ __launch_bounds__(256) void pe_zero_kernel(float4* __restrict__ p, int n4) {
  int stride = gridDim.x * blockDim.x;
  for (int i = blockIdx.x * blockDim.x + threadIdx.x; i < n4; i += stride)
    p[i] = make_float4(0.f, 0.f, 0.f, 0.f);
}

// ---------------------------------------------------------------------------
// K1: edge featurize (concat + cos time-encode) -> GEMM (K=272,N=128) -> scatter
// 32 edges per block, 8 waves each owning a 16-col tile; A tile in LDS.
// ---------------------------------------------------------------------------
__global__ __launch_bounds__(256) void pe_edge_kernel(
    const float* __restrict__ ef, const float* __restrict__ ts,
    const float* __restrict__ fw, const float* __restrict__ fb,
    const int* __restrict__ inds, float* __restrict__ xscat) {
  __shared__ float sA[32 * 276];  // 32 rows x 272 cols, stride 276 (bank-safe)
  __shared__ int sI[32];
  const int tid = threadIdx.x;
  const int ebase = blockIdx.x * 32;
  if (tid < 32) sI[tid] = inds[ebase + tid];
  {
    const int row = tid >> 3, c0 = tid & 7;
    const float tv = ts[ebase + row];
    const float* efr = ef + (size_t)(ebase + row) * 172;
    for (int c = c0; c < 272; c += 8) {
      float v;
      if (c < 172) {
        v = efr[c];
      } else {
        // freq[j] = 10^(-9j/99) = exp(-j * 9*ln(10)/99)
        const float fr = __expf((float)(c - 172) * (-0.2093259176f));
        v = cosf(tv * fr);
      }
      sA[row * 276 + c] = v;
    }
  }
  __syncthreads();
  const int wave = tid >> 5, lane = tid & 31;
  const int half = lane >> 4, l16 = lane & 15;
  const int col = wave * 16 + l16;  // N=128, always valid
  const float* wk = fw + (size_t)col * 272 + 2 * half;
  const float* ak = &sA[l16 * 276 + 2 * half];
  v8f acc0 = {0.f, 0.f, 0.f, 0.f, 0.f, 0.f, 0.f, 0.f};
  v8f acc1 = acc0;
#pragma unroll 4
  for (int kk = 0; kk < 272; kk += 4) {
    const v2f bf = *(const v2f*)(wk + kk);
    const v2f a0 = *(const v2f*)(ak + kk);
    acc0 = wmma4(a0, bf, acc0);
    const v2f a1 = *(const v2f*)(ak + kk + 16 * 276);
    acc1 = wmma4(a1, bf, acc1);
  }
  const float bv = fb[col];
#pragma unroll
  for (int r = 0; r < 8; ++r) {
    const int r0 = half * 8 + r;
    xscat[(size_t)sI[r0] * 128 + col] = acc0[r] + bv;
    xscat[(size_t)sI[16 + r0] * 128 + col] = acc1[r] + bv;
  }
}

// ---------------------------------------------------------------------------
// Generic WMMA GEMM: C(M x N) = A(M x K) @ W(N x K)^T  [+ epilogue]
// Block = 64 rows x 128 cols (8 waves x 16 cols).
// A is double-buffered in LDS via async global->LDS copies (ASYNCcnt).
// ---------------------------------------------------------------------------
enum { EPI_NONE = 0, EPI_BIAS = 1, EPI_BIAS_PE = 2, EPI_RESID = 3, EPI_SOFTPLUS = 4 };

template <int EPI>
__global__ __launch_bounds__(256) void pe_gemm_kernel(
    const float* __restrict__ A, int lda, const float* __restrict__ W, int ldw,
    float* __restrict__ C, int ldc, const float* __restrict__ bias, int N, int K) {
  __shared__ float sA[2][64 * 36];  // 2 x (64 rows x 32-K chunk), stride 36
  const int tid = threadIdx.x;
  const int mbase = blockIdx.x * 64;
  const int wave = tid >> 5, lane = tid & 31;
  const int half = lane >> 4, l16 = lane & 15;
  const int col = blockIdx.y * 128 + wave * 16 + l16;
  const bool valid = col < N;
  const int bcol = valid ? col : (N - 1);
  const float* wrow = W + (size_t)bcol * ldw + 2 * half;

  // Per-thread staging slots: two b128 copies per chunk (rows row0 and row0+32)
  const int row0 = tid >> 3;        // 0..31
  const int cg0 = (tid & 7) * 4;    // 0,4,...,28
  const unsigned lbase0 = (unsigned)(unsigned long long)(as3f*)&sA[0][0];
  const unsigned lbase1 = (unsigned)(unsigned long long)(as3f*)&sA[1][0];
  const unsigned loff_a = (unsigned)(row0 * 36 + cg0) * 4u;
  const unsigned loff_b = (unsigned)((row0 + 32) * 36 + cg0) * 4u;

  auto stage = [&](int kb, int buf) {
    const int kc = (K - kb < 32) ? (K - kb) : 32;
    if (cg0 < kc) {
      const unsigned lb = buf ? lbase1 : lbase0;
      async_ld_b128(lb + loff_a, A + (size_t)(mbase + row0) * lda + kb + cg0);
      async_ld_b128(lb + loff_b, A + (size_t)(mbase + row0 + 32) * lda + kb + cg0);
    }
  };

  v8f acc[4];
#pragma unroll
  for (int j = 0; j < 4; ++j) acc[j] = (v8f){0.f, 0.f, 0.f, 0.f, 0.f, 0.f, 0.f, 0.f};

  stage(0, 0);
  int buf = 0;
  for (int kb = 0; kb < K; kb += 32) {
    const int kc = (K - kb < 32) ? (K - kb) : 32;
    const bool pre = (kb + 32) < K;
    if (pre) {
      stage(kb + 32, buf ^ 1);
      // prev chunk (issued earlier, in-order completion) done when <= 2 remain
      asm volatile("s_wait_asynccnt 0x2" ::: "memory");
    } else {
      asm volatile("s_wait_asynccnt 0x0" ::: "memory");
    }
    __syncthreads();
    const float* ak = &sA[buf][l16 * 36 + 2 * half];
#pragma unroll 4
    for (int kk = 0; kk < kc; kk += 4) {
      const v2f bf = *(const v2f*)(wrow + kb + kk);
#pragma unroll
      for (int j = 0; j < 4; ++j) {
        const v2f af = *(const v2f*)(ak + kk + j * 16 * 36);
        acc[j] = wmma4(af, bf, acc[j]);
      }
    }
    __syncthreads();
    buf ^= 1;
  }

  float bval = 0.f;
  if ((EPI == EPI_BIAS || EPI == EPI_BIAS_PE || EPI == EPI_SOFTPLUS) && valid)
    bval = bias[col];
  float invf = 0.f;
  if (EPI == EPI_BIAS_PE)  // inv_freq = 10000^(-(col&~1)/128)
    invf = __expf((float)(col & ~1) * (-9.210340372f / 128.f));

#pragma unroll
  for (int j = 0; j < 4; ++j) {
#pragma unroll
    for (int r = 0; r < 8; ++r) {
      if (valid) {
        const int row = mbase + j * 16 + half * 8 + r;
        const size_t idx = (size_t)row * ldc + col;
        float v = acc[j][r];
        if (EPI == EPI_BIAS) {
          v += bval;
        } else if (EPI == EPI_BIAS_PE) {
          const float arg = (float)(row & 31) * invf;
          v += bval + ((col & 1) ? cosf(arg) : sinf(arg));
        } else if (EPI == EPI_RESID) {
          v += C[idx];
        } else if (EPI == EPI_SOFTPLUS) {
          const float x = v + bval;
          v = (x > 20.f) ? x : log1pf(__expf(x));
        }
        C[idx] = v;
      }
    }
  }
}

// ---------------------------------------------------------------------------
// RMSNorm over rows of 128 (one wave32 per row, float4 per lane)
// ---------------------------------------------------------------------------
__global__ __launch_bounds__(256) void pe_rmsnorm_kernel(
    const float* __restrict__ x, const float* __restrict__ w, float* __restrict__ out) {
  const int lane = threadIdx.x & 31, wave = threadIdx.x >> 5;
  const size_t row = (size_t)blockIdx.x * 8 + wave;
  const float4 v = *(const float4*)(x + row * 128 + lane * 4);
  float ss = v.x * v.x + v.y * v.y + v.z * v.z + v.w * v.w;
#pragma unroll
  for (int off = 16; off > 0; off >>= 1) ss += __shfl_xor(ss, off, 32);
  const float sc = rsqrtf(ss * (1.f / 128.f) + EPS_RMS);
  const float4 wv = *(const float4*)(w + lane * 4);
  float4 o;
  o.x = v.x * sc * wv.x; o.y = v.y * sc * wv.y;
  o.z = v.z * sc * wv.z; o.w = v.w * sc * wv.w;
  *(float4*)(out + row * 128 + lane * 4) = o;
}

// ---------------------------------------------------------------------------
// Depthwise causal conv1d (k=4) + SiLU. One block per graph, thread=channel.
// ---------------------------------------------------------------------------
__global__ __launch_bounds__(256) void pe_conv_silu_kernel(
    const float* __restrict__ xz, const float* __restrict__ cw,
    const float* __restrict__ cb, float* __restrict__ xsc) {
  const int b = blockIdx.x, d = threadIdx.x;
  const float w0 = cw[d * 4 + 0], w1 = cw[d * 4 + 1];
  const float w2 = cw[d * 4 + 2], w3 = cw[d * 4 + 3];
  const float bv = cb[d];
  float h0 = 0.f, h1 = 0.f, h2 = 0.f;
  const size_t rb = (size_t)b * 32;
  for (int l = 0; l < 32; ++l) {
    const float xc = xz[(rb + l) * 512 + d];  // xs half of xz
    const float a = h0 * w0 + h1 * w1 + h2 * w2 + xc * w3 + bv;
    xsc[(rb + l) * 256 + d] = a / (1.f + __expf(-a));  // silu
    h0 = h1; h1 = h2; h2 = xc;
  }
}

// ---------------------------------------------------------------------------
// Selective scan: thread = (graph b, channel d), 32-state recurrence in regs,
// B/C staged in LDS (broadcast). Fuses D-skip and silu(res) gating.
// ---------------------------------------------------------------------------
__global__ __launch_bounds__(256) void pe_scan_kernel(
    const float* __restrict__ proj, const float* __restrict__ dtb,
    const float* __restrict__ xsc, const float* __restrict__ xz,
    const float* __restrict__ Alog, const float* __restrict__ Dsk,
    float* __restrict__ y) {
  __shared__ float Bs[1024], Cs[1024];
  const int b = blockIdx.x, d = threadIdx.x;
  for (int i = threadIdx.x; i < 1024; i += 256) {
    const int l = i >> 5, s = i & 31;
    const size_t pr = (size_t)(b * 32 + l) * 80;
    Bs[i] = proj[pr + 8 + s];
    Cs[i] = proj[pr + 40 + s];
  }
  __syncthreads();
  float nA[32], h[32];
#pragma unroll
  for (int s = 0; s < 32; ++s) {
    nA[s] = -__expf(Alog[d * 32 + s]);
    h[s] = 0.f;
  }
  const float Dv = Dsk[d];
  for (int l = 0; l < 32; ++l) {
    const size_t r = (size_t)(b * 32 + l);
    const float dt = dtb[r * 256 + d];
    const float xv = xsc[r * 256 + d];
    const float res = xz[r * 512 + 256 + d];
    const float dtx = dt * xv;
    float acc = 0.f;
    const float* Bl = &Bs[l * 32];
    const float* Cl = &Cs[l * 32];
#pragma unroll
    for (int s = 0; s < 32; ++s) {
      h[s] = __expf(dt * nA[s]) * h[s] + dtx * Bl[s];
      acc += h[s] * Cl[s];
    }
    const float val = acc + xv * Dv;
    const float sr = res / (1.f + __expf(-res));
    y[r * 256 + d] = val * sr;
  }
}

// ---------------------------------------------------------------------------
// Final RMSNorm + mean over L + head matvec. One block per graph.
// ---------------------------------------------------------------------------
__global__ __launch_bounds__(256) void pe_head_kernel(
    const float* __restrict__ x, const float* __restrict__ rw,
    const float* __restrict__ hw, const float* __restrict__ hb,
    float* __restrict__ out) {
  __shared__ float m[128];
  const int b = blockIdx.x, tid = threadIdx.x;
  const int lane = tid & 31, wave = tid >> 5;
  if (tid < 128) m[tid] = 0.f;
  __syncthreads();
  const float4 wv = *(const float4*)(rw + lane * 4);
  for (int it = 0; it < 4; ++it) {
    const int l = it * 8 + wave;
    const float4 v = *(const float4*)(x + ((size_t)b * 32 + l) * 128 + lane * 4);
    float ss = v.x * v.x + v.y * v.y + v.z * v.z + v.w * v.w;
#pragma unroll
    for (int off = 16; off > 0; off >>= 1) ss += __shfl_xor(ss, off, 32);
    const float sc = rsqrtf(ss * (1.f / 128.f) + EPS_RMS);
    atomicAdd(&m[lane * 4 + 0], v.x * sc * wv.x);
    atomicAdd(&m[lane * 4 + 1], v.y * sc * wv.y);
    atomicAdd(&m[lane * 4 + 2], v.z * sc * wv.z);
    atomicAdd(&m[lane * 4 + 3], v.w * sc * wv.w);
  }
  __syncthreads();
  if (tid < 128) m[tid] *= (1.f / 32.f);
  __syncthreads();
  if (tid < 128) {
    float acc = hb[tid];
    const float* hr = hw + (size_t)tid * 128;
    for (int i = 0; i < 128; ++i) acc += m[i] * hr[i];
    out[(size_t)b * 128 + tid] = acc;
  }
}

// ---------------------------------------------------------------------------
extern "C" void kernel_launch(void* const* d_in, const int* in_sizes, int n_in,
                              void* d_out, int out_size, void* d_ws, size_t ws_size,
                              hipStream_t stream) {
  (void)in_sizes; (void)n_in; (void)out_size; (void)ws_size;
  const float* edge_feats = (const float*)d_in[0];
  const float* edge_ts    = (const float*)d_in[1];
  const float* feat_w     = (const float*)d_in[2];
  const float* feat_b     = (const float*)d_in[3];
  const float* pad_w      = (const float*)d_in[4];
  const float* pad_b      = (const float*)d_in[5];
  const float* head_w     = (const float*)d_in[6];
  const float* head_b     = (const float*)d_in[7];
  const float* rms_w      = (const float*)d_in[8];
  const float* norm_w     = (const float*)d_in[9];
  const float* in_proj_w  = (const float*)d_in[10];
  const float* conv_w     = (const float*)d_in[11];
  const float* conv_b     = (const float*)d_in[12];
  const float* x_proj_w   = (const float*)d_in[13];
  const float* dt_proj_w  = (const float*)d_in[14];
  const float* dt_proj_b  = (const float*)d_in[15];
  const float* A_log      = (const float*)d_in[16];
  const float* D_skip     = (const float*)d_in[17];
  const float* out_proj_w = (const float*)d_in[18];
  const int*   inds       = (const int*)d_in[20];

  // Workspace layout (floats). Peak ~436 MB with aliasing:
  float* ws    = (float*)d_ws;
  float* xscat = ws;                      // 67,108,864  (2048*256*128) scatter buf
  float* xres  = ws + 67108864ull;        //  8,388,608  (65536*128) residual stream
  float* dtb   = ws + 75497472ull;        // 16,777,216  (65536*256) dt
  float* ybuf  = ws + 92274688ull;        // 16,777,216  (65536*256) scan output
  // Per-layer scratch aliases inside xscat (free after pad GEMM):
  float* xn   = xscat;                    //  8,388,608  rmsnorm'd input
  float* xz   = xscat + 8388608ull;       // 33,554,432  in_proj output (xs|res)
  float* xsc  = xscat + 41943040ull;      // 16,777,216  conv+silu output
  float* proj = xscat + 58720256ull;      //  5,242,880  x_proj output (ld=80)

  // Stage 1: zero slots, edge featurize+GEMM+scatter, pad GEMM (+bias+PE)
  pe_zero_kernel<<<2048, 256, 0, stream>>>((float4*)xscat, 16777216);
  pe_edge_kernel<<<8192, 256, 0, stream>>>(edge_feats, edge_ts, feat_w, feat_b,
                                           inds, xscat);
  pe_gemm_kernel<EPI_BIAS_PE><<<dim3(1024, 1), 256, 0, stream>>>(
      xscat, 1024, pad_w, 1024, xres, 128, pad_b, 128, 1024);

  // Stage 2: two Mamba layers with residual
  for (int layer = 0; layer < 2; ++layer) {
    pe_rmsnorm_kernel<<<8192, 256, 0, stream>>>(xres, norm_w + layer * 128, xn);
    pe_gemm_kernel<EPI_NONE><<<dim3(1024, 4), 256, 0, stream>>>(
        xn, 128, in_proj_w + (size_t)layer * 512 * 128, 128, xz, 512, nullptr,
        512, 128);
    pe_conv_silu_kernel<<<2048, 256, 0, stream>>>(
        xz, conv_w + layer * 256 * 4, conv_b + layer * 256, xsc);
    pe_gemm_kernel<EPI_NONE><<<dim3(1024, 1), 256, 0, stream>>>(
        xsc, 256, x_proj_w + (size_t)layer * 72 * 256, 256, proj, 80, nullptr,
        72, 256);
    pe_gemm_kernel<EPI_SOFTPLUS><<<dim3(1024, 2), 256, 0, stream>>>(
        proj, 80, dt_proj_w + (size_t)layer * 256 * 8, 8, dtb, 256,
        dt_proj_b + layer * 256, 256, 8);
    pe_scan_kernel<<<2048, 256, 0, stream>>>(
        proj, dtb, xsc, xz, A_log + (size_t)layer * 256 * 32,
        D_skip + layer * 256, ybuf);
    pe_gemm_kernel<EPI_RESID><<<dim3(1024, 1), 256, 0, stream>>>(
        ybuf, 256, out_proj_w + (size_t)layer * 128 * 256, 256, xres, 128,
        nullptr, 128, 256);
  }

  // Stage 3: final norm + mean + head
  pe_head_kernel<<<2048, 256, 0, stream>>>(xres, rms_w, head_w, head_b,
                                           (float*)d_out);
}